// SegmentNormalizeLayer_67551245631958
// MI455X (gfx1250) — compile-verified
//
#include <hip/hip_runtime.h>
#include <hip/hip_bf16.h>

// SegmentNormalize for MI455X (gfx1250).
// Bandwidth-bound: ~776MB total traffic -> ~33us floor at 23.3 TB/s.
// Pass 1 computes the masks@x / masks@x^2 einsums with V_WMMA_F32_16X16X4_F32
// (16 channels x 16 classes, K=4 pixels per WMMA, fp32 accumulate), fed by
// double-buffered GLOBAL_LOAD_ASYNC_TO_LDS_B128 staging (ASYNCcnt-tracked).

typedef __attribute__((ext_vector_type(2))) float v2f;
typedef __attribute__((ext_vector_type(8))) float v8f;

#define SN_B 4
#define SN_C 64
#define SN_HW (512 * 512)
#define SN_K 16

constexpr int TILE      = 256;        // pixels staged per tile
constexpr int XSTRIDE   = TILE + 4;   // dword stride pad -> conflict-free b64 A reads
constexpr int NWAVES    = 8;          // 256 threads, wave32
constexpr int PIXBLOCKS = 32;         // pixel-blocks per (b, channel-group)
constexpr int PIX_PER_BLOCK = SN_HW / PIXBLOCKS;      // 8192
constexpr int TILES_PER_BLOCK = PIX_PER_BLOCK / TILE; // 32

// ---------------------------------------------------------------- zero ws ---
__global__ __launch_bounds__(256) void sn_zero_ws(float* __restrict__ ws) {
    int i = blockIdx.x * 256 + threadIdx.x;
    if (i < 2 * SN_B * SN_C * SN_K) ws[i] = 0.0f;   // s1 + s2 regions
}

// --------------------------------------------------- pass 1: WMMA stats -----
// grid = (PIXBLOCKS, B * C/16), block = 256
__global__ __launch_bounds__(256) void sn_stats_wmma(
    const float* __restrict__ x, const int* __restrict__ seg,
    float* __restrict__ s1g, float* __restrict__ s2g)
{
    __shared__ __align__(16) float xs[2][16 * XSTRIDE]; // double-buffered x tiles
    __shared__ __align__(16) int   ss[2][TILE];         // double-buffered seg tiles
    __shared__ float red[NWAVES * 512];                 // cross-wave reduction

    const int tid    = threadIdx.x;
    const int lane   = tid & 31;
    const int wave   = tid >> 5;
    const int lane15 = lane & 15;        // = M (channel) for A, = N (class) for B
    const int laneHi = lane >> 4;        // K group: lanes 0-15 -> K0,K1; 16-31 -> K2,K3

    const int combo = blockIdx.y;        // 0..15
    const int b     = combo >> 2;        // batch
    const int cg    = combo & 3;         // channel group (16 channels each)

    const int pix0 = blockIdx.x * PIX_PER_BLOCK;
    const float* xb = x   + (size_t)(b * SN_C + cg * 16) * SN_HW;
    const int*   sb = seg + (size_t)b * SN_HW;

    // per-thread static addressing for the async staging (5 ops / tile):
    //   4 x float4 rows of the 16ch x 256px tile + 1 seg int
    const int i0  = tid;                 // float4 unit index base, rounds add 256
    // issue one tile's async loads into buffer `buf`
    auto issue_tile = [&](int t, int buf) {
        const int tbase = pix0 + t * TILE;
        {   // seg: 1 int per thread
            unsigned lds = (unsigned)(size_t)&ss[buf][tid];
            unsigned go  = (unsigned)((tbase + tid) * 4);
            asm volatile("global_load_async_to_lds_b32 %0, %1, %2"
                         :: "v"(lds), "v"(go), "s"(sb) : "memory");
        }
        #pragma unroll
        for (int r = 0; r < 4; ++r) {
            const int i  = r * 256 + i0;     // float4 unit index in [0,1024)
            const int ch = i >> 6;           // 64 float4 per channel row
            const int po = (i & 63) << 2;    // dword offset in row
            unsigned lds = (unsigned)(size_t)&xs[buf][ch * XSTRIDE + po];
            unsigned go  = (unsigned)(((unsigned)ch * SN_HW + tbase + po) * 4);
            asm volatile("global_load_async_to_lds_b128 %0, %1, %2"
                         :: "v"(lds), "v"(go), "s"(xb) : "memory");
        }
    };

    v8f acc1 = {};  // sum(x)   : 16ch x 16class
    v8f acc2 = {};  // sum(x*x)

    issue_tile(0, 0);                    // prologue: stage tile 0

    for (int t = 0; t < TILES_PER_BLOCK; ++t) {
        const int buf = t & 1;
        if (t + 1 < TILES_PER_BLOCK) {
            issue_tile(t + 1, buf ^ 1);  // prefetch next tile into other buffer
            // 5 newest (next tile) may stay outstanding; 5 oldest (this tile) done
            asm volatile("s_wait_asynccnt 0x5" ::: "memory");
        } else {
            asm volatile("s_wait_asynccnt 0x0" ::: "memory");
        }
        __syncthreads();                 // all waves' tile-t data visible in LDS

        // each wave consumes 32 pixels of the tile, 4 per WMMA
        #pragma unroll
        for (int c4 = 0; c4 < 8; ++c4) {
            const int p = wave * 32 + c4 * 4 + laneHi * 2;  // this lane's pixel pair
            const float a0 = xs[buf][lane15 * XSTRIDE + p];
            const float a1 = xs[buf][lane15 * XSTRIDE + p + 1];
            const int  sgA = ss[buf][p];
            const int  sgB = ss[buf][p + 1];

            v2f a;  a.x  = a0;      a.y  = a1;
            v2f aq; aq.x = a0 * a0; aq.y = a1 * a1;
            v2f bm; bm.x = (sgA == lane15) ? 1.0f : 0.0f;   // one-hot column
                    bm.y = (sgB == lane15) ? 1.0f : 0.0f;

            acc1 = __builtin_amdgcn_wmma_f32_16x16x4_f32(
                false, a,  false, bm, (short)0, acc1, false, false);
            acc2 = __builtin_amdgcn_wmma_f32_16x16x4_f32(
                false, aq, false, bm, (short)0, acc2, false, false);
        }
        __syncthreads();                 // done reading buf before it is refilled
    }

    // cross-wave reduce the 16x16 accumulators, then one atomic flush
    #pragma unroll
    for (int j = 0; j < 8; ++j) {
        red[wave * 512 +       j * 32 + lane] = acc1[j];
        red[wave * 512 + 256 + j * 32 + lane] = acc2[j];
    }
    __syncthreads();

    for (int e = tid; e < 512; e += 256) {
        float s = 0.0f;
        #pragma unroll
        for (int w = 0; w < NWAVES; ++w) s += red[w * 512 + e];
        const int stat = e >> 8;          // 0 -> s1, 1 -> s2
        const int j    = (e >> 5) & 7;    // D VGPR index
        const int ln   = e & 31;          // D lane
        const int c    = cg * 16 + ((ln < 16) ? j : (j + 8));  // ISA C/D layout
        const int k    = ln & 15;
        float* dst = stat ? s2g : s1g;
        atomicAdd(&dst[((b * SN_C + c) * SN_K) + k], s);
    }
}

// ------------------------------------------- pass 1b: tiny per-(b,c) math ---
__global__ __launch_bounds__(256) void sn_finalize_stats(
    const float* __restrict__ s1, const float* __restrict__ s2,
    float* __restrict__ inv, float* __restrict__ constv)
{
    const int bc = threadIdx.x;          // one thread per (b,c), 256 total
    constexpr float rHW = 1.0f / (float)SN_HW;
    float cst = 0.0f;
    #pragma unroll
    for (int k = 0; k < SN_K; ++k) {
        const float m  = s1[bc * SN_K + k] * rHW;
        const float v  = fmaxf(s2[bc * SN_K + k] * rHW - m * m, 0.0f);
        const float iv = rsqrtf(v + 1e-5f);
        inv[bc * SN_K + k] = iv;
        cst += m * iv;
    }
    constv[bc] = cst;
}

// -------------------------------------------------- pass 2: apply (stream) --
// grid = (128, B*C), block = 256; float4 per thread, 2 iters
__global__ __launch_bounds__(256) void sn_apply(
    const float* __restrict__ x, const int* __restrict__ seg,
    const float* __restrict__ inv, const float* __restrict__ constv,
    float* __restrict__ out)
{
    __shared__ float s_inv[SN_K];
    __shared__ float s_c;
    const int bc = blockIdx.y;
    const int b  = bc >> 6;
    if (threadIdx.x < SN_K) s_inv[threadIdx.x] = inv[bc * SN_K + threadIdx.x];
    if (threadIdx.x == 0)   s_c = constv[bc];
    __syncthreads();

    constexpr int V4 = SN_HW / 4;        // 65536 float4 per (b,c)
    const float4* xv = (const float4*)x   + (size_t)bc * V4;
    const int4*   sv = (const int4*)seg   + (size_t)b  * V4;
    float4*       ov = (float4*)out       + (size_t)bc * V4;
    const float cst = s_c;

    const int base = blockIdx.x * 512;
    #pragma unroll
    for (int it = 0; it < 2; ++it) {
        const int idx = base + it * 256 + threadIdx.x;
        const float4 xd = xv[idx];
        const int4   sd = sv[idx];
        float4 o;
        o.x = xd.x * s_inv[sd.x] - cst;
        o.y = xd.y * s_inv[sd.y] - cst;
        o.z = xd.z * s_inv[sd.z] - cst;
        o.w = xd.w * s_inv[sd.w] - cst;
        ov[idx] = o;
    }
}

// ----------------------------------------------------------------- launch ---
extern "C" void kernel_launch(void* const* d_in, const int* in_sizes, int n_in,
                              void* d_out, int out_size, void* d_ws, size_t ws_size,
                              hipStream_t stream) {
    (void)in_sizes; (void)n_in; (void)out_size; (void)ws_size;
    const float* x   = (const float*)d_in[0];
    const int*   seg = (const int*)d_in[1];
    float*       out = (float*)d_out;

    float* ws     = (float*)d_ws;
    float* s1     = ws;                       // 4096
    float* s2     = ws + 4096;                // 4096
    float* inv    = ws + 8192;                // 4096
    float* constv = ws + 12288;               // 256

    sn_zero_ws<<<dim3(32), 256, 0, stream>>>(ws);
    sn_stats_wmma<<<dim3(PIXBLOCKS, SN_B * (SN_C / 16)), 256, 0, stream>>>(x, seg, s1, s2);
    sn_finalize_stats<<<dim3(1), 256, 0, stream>>>(s1, s2, inv, constv);
    sn_apply<<<dim3(128, SN_B * SN_C), 256, 0, stream>>>(x, seg, inv, constv, out);
}